// SBSmyrfAttention_47304769798227
// MI455X (gfx1250) — compile-verified
//
#include <hip/hip_runtime.h>
#include <math.h>

// ---------------- problem constants (from reference setup_inputs) ------------
#define TSEQ   4096
#define NB     32      // buckets per (hash, bh):  TSEQ / BKT
#define BKT    128     // Q_CLUSTER == K_CLUSTER
#define BATCH  2
#define HEADS  8
#define BH     16      // BATCH*HEADS
#define EDIM   64
#define DDIM   64
#define MFEAT  256
#define NHASH  4

#define TEMP   0.125f                 // 1/sqrt(E)
#define DNF    0.3535533905932738f    // sqrt(TEMP)
#define RATIO  0.0625f                // MFEAT^-0.5
#define DIAGC  0.0625f                // 0.5*TEMP

typedef float v2f __attribute__((ext_vector_type(2)));
typedef float v8f __attribute__((ext_vector_type(8)));

// D(16x16,f32) += A(16x4,f32) * B(4x16,f32)
// A frag: lane%16 = M row; lanes<16 hold K={0,1}, lanes>=16 hold K={2,3}
// B frag: lane%16 = N col; lanes<16 hold K={0,1}, lanes>=16 hold K={2,3}
// C/D:    VGPR r: lanes0-15 -> (M=r, N=lane), lanes16-31 -> (M=r+8, N=lane-16)
__device__ __forceinline__ v8f wmma4(float ax, float ay, float bx, float by, v8f c) {
  v2f a = {ax, ay};
  v2f b = {bx, by};
  return __builtin_amdgcn_wmma_f32_16x16x4_f32(false, a, false, b, (short)0, c,
                                               false, false);
}

// order-preserving float<->uint encoding for signed atomic max
__device__ __forceinline__ unsigned fenc(float f) {
  unsigned u = __float_as_uint(f);
  return u ^ ((unsigned)((int)u >> 31) | 0x80000000u);
}
__device__ __forceinline__ float fdec(unsigned u) {
  unsigned v = (u & 0x80000000u) ? (u ^ 0x80000000u) : ~u;
  return __uint_as_float(v);
}

// ---------------------------------------------------------------------------
__global__ void k_init(unsigned* mq2, unsigned* mk2, unsigned* kmax) {
  int i = threadIdx.x;
  if (i < BH) { mq2[i] = 0u; mk2[i] = 0u; kmax[i] = 0u; }
}

// per-token squared norms + per-bh max (norms are >=0 -> raw-bit atomicMax ok)
__global__ void k_norms(const float* __restrict__ q, const float* __restrict__ k,
                        float* qn2, float* kn2, unsigned* mq2, unsigned* mk2) {
  int gid = blockIdx.x * blockDim.x + threadIdx.x;   // over BH*TSEQ
  int bh = gid / TSEQ, t = gid % TSEQ;
  int b = bh / HEADS, hh = bh % HEADS;
  size_t base = ((size_t)(b * TSEQ + t) * HEADS + hh) * EDIM;
  float sq = 0.f, sk = 0.f;
  for (int e = 0; e < EDIM; ++e) {
    float a = q[base + e]; sq += a * a;
    float c = k[base + e]; sk += c * c;
  }
  qn2[gid] = sq; kn2[gid] = sk;
  __shared__ float smq[256], smk[256];
  smq[threadIdx.x] = sq; smk[threadIdx.x] = sk;
  __syncthreads();
  for (int s = 128; s > 0; s >>= 1) {
    if (threadIdx.x < s) {
      smq[threadIdx.x] = fmaxf(smq[threadIdx.x], smq[threadIdx.x + s]);
      smk[threadIdx.x] = fmaxf(smk[threadIdx.x], smk[threadIdx.x + s]);
    }
    __syncthreads();
  }
  if (threadIdx.x == 0) {
    atomicMax(&mq2[bh], __float_as_uint(smq[0]));
    atomicMax(&mk2[bh], __float_as_uint(smk[0]));
  }
}

// XBOX+ transform + E2LSH hashes
__global__ void k_hash(const float* __restrict__ q, const float* __restrict__ k,
                       const float* __restrict__ alpha, const float* __restrict__ beta,
                       const float* qn2, const float* kn2,
                       const unsigned* mq2, const unsigned* mk2,
                       float* qhash, float* khash) {
  int gid = blockIdx.x * blockDim.x + threadIdx.x;   // over BH*TSEQ
  int bh = gid / TSEQ, t = gid % TSEQ;
  int b = bh / HEADS, hh = bh % HEADS;
  size_t base = ((size_t)(b * TSEQ + t) * HEADS + hh) * EDIM;
  float MQ2 = __uint_as_float(mq2[bh]), MK2 = __uint_as_float(mk2[bh]);
  float extq = sqrtf(fmaxf(MQ2 + MK2 - qn2[gid], 0.f));
  float extk = sqrtf(fmaxf(MQ2 + MK2 - kn2[gid], 0.f));
  float sq[NHASH] = {0.f, 0.f, 0.f, 0.f};
  float sk[NHASH] = {0.f, 0.f, 0.f, 0.f};
  for (int e = 0; e < EDIM; ++e) {
    float qa = q[base + e], ka = k[base + e];
    for (int j = 0; j < NHASH; ++j) {
      float al = alpha[e * NHASH + j];
      sq[j] += qa * al; sk[j] += ka * al;
    }
  }
  for (int j = 0; j < NHASH; ++j) {
    qhash[((size_t)j * BH + bh) * TSEQ + t] = sq[j] + extq * alpha[64 * NHASH + j] + beta[j];
    khash[((size_t)j * BH + bh) * TSEQ + t] = sk[j] + extk * alpha[65 * NHASH + j] + beta[j];
  }
}

// bitonic argsort (ascending) of one length-4096 row per block; emits perm+inv
__global__ void __launch_bounds__(512) k_sort(const float* __restrict__ hash,
                                              int* pos, int* rev) {
  __shared__ float key[TSEQ];
  __shared__ int   val[TSEQ];
  const float* h = hash + (size_t)blockIdx.x * TSEQ;
  for (int i = threadIdx.x; i < TSEQ; i += blockDim.x) { key[i] = h[i]; val[i] = i; }
  __syncthreads();
  for (int k = 2; k <= TSEQ; k <<= 1) {
    for (int j = k >> 1; j > 0; j >>= 1) {
      for (int i = threadIdx.x; i < TSEQ; i += blockDim.x) {
        int ixj = i ^ j;
        if (ixj > i) {
          bool up = ((i & k) == 0);
          float ki = key[i], kj = key[ixj];
          if (up ? (ki > kj) : (ki < kj)) {
            key[i] = kj; key[ixj] = ki;
            int tv = val[i]; val[i] = val[ixj]; val[ixj] = tv;
          }
        }
      }
      __syncthreads();
    }
  }
  int* p = pos + (size_t)blockIdx.x * TSEQ;
  int* r = rev + (size_t)blockIdx.x * TSEQ;
  for (int i = threadIdx.x; i < TSEQ; i += blockDim.x) { p[i] = val[i]; r[val[i]] = i; }
}

// logv = dn * X @ proj^T - 0.5*dn^2*||x||^2   (WMMA; z=0 -> query, z=1 -> key)
__global__ void __launch_bounds__(512) k_logv(const float* __restrict__ q,
                                              const float* __restrict__ k,
                                              const float* __restrict__ proj,
                                              const float* qn2, const float* kn2,
                                              float* lq, float* lk, unsigned* kmaxb) {
  int isK = blockIdx.z;
  const float* x  = isK ? k : q;
  const float* n2 = isK ? kn2 : qn2;
  float* dst      = isK ? lk : lq;
  int bh = blockIdx.y;
  int b = bh / HEADS, hh = bh % HEADS;
  int lane = threadIdx.x & 31;
  int wave = threadIdx.x >> 5;          // 0..15 -> feature column tile
  int l16 = lane & 15;
  int kh  = (lane >> 4) << 1;
  int hi8 = (lane >> 4) << 3;
  int m0 = blockIdx.x * 16;             // token rows
  int n0 = wave * 16;                   // feature cols
  size_t abase = ((size_t)(b * TSEQ + (m0 + l16)) * HEADS + hh) * EDIM;
  v8f acc = {};
  for (int k0 = 0; k0 < EDIM; k0 += 4) {
    float ax = x[abase + k0 + kh];
    float ay = x[abase + k0 + kh + 1];
    float bx = proj[(size_t)(n0 + l16) * EDIM + k0 + kh];
    float by = proj[(size_t)(n0 + l16) * EDIM + k0 + kh + 1];
    acc = wmma4(ax, ay, bx, by, acc);
  }
  float lmax = -3.4e38f;
  for (int rr = 0; rr < 8; ++rr) {
    int row = m0 + rr + hi8;
    float lv = DNF * acc[rr] - DIAGC * n2[(size_t)bh * TSEQ + row];
    dst[((size_t)bh * TSEQ + row) * MFEAT + n0 + l16] = lv;
    lmax = fmaxf(lmax, lv);
  }
  if (isK) {
    for (int off = 1; off < 32; off <<= 1)
      lmax = fmaxf(lmax, __shfl_xor(lmax, off, 32));
    if (lane == 0) atomicMax(&kmaxb[bh], fenc(lmax));
  }
}

// per-row stabilizer for q; prime_log_scale = q_ls + k_ls
__global__ void k_qls(const float* __restrict__ lq, const unsigned* kmaxb,
                      float* qls, float* pls) {
  int gid = blockIdx.x * blockDim.x + threadIdx.x;   // over BH*TSEQ
  int bh = gid / TSEQ;
  const float* row = lq + (size_t)gid * MFEAT;
  float m = -3.4e38f;
  for (int j = 0; j < MFEAT; ++j) m = fmaxf(m, row[j]);
  qls[gid] = m;
  pls[gid] = m + fdec(kmaxb[bh]);
}

// q' = exp(logv - q_ls)*ratio ; k' = exp(logv - k_ls)*ratio  (in place)
__global__ void k_prime(float* qp, float* kp, const float* qls, const unsigned* kmaxb) {
  size_t idx = (size_t)blockIdx.x * blockDim.x + threadIdx.x;  // over BH*TSEQ*MFEAT
  size_t bt = idx / MFEAT;
  int bh = (int)(bt / TSEQ);
  qp[idx] = expf(qp[idx] - qls[bt]) * RATIO;
  kp[idx] = expf(kp[idx] - fdec(kmaxb[bh])) * RATIO;
}

__global__ void k_ksum(const float* __restrict__ kp, float* ksum) {
  int bh = blockIdx.x, m = threadIdx.x;                // 16 blocks x 256 threads
  float s = 0.f;
  for (int t = 0; t < TSEQ; ++t) s += kp[((size_t)bh * TSEQ + t) * MFEAT + m];
  ksum[bh * MFEAT + m] = s;
}

// kv[bh][m][d] = sum_s k'[s][m] * v[s][d]    (WMMA, K = 4096)
__global__ void k_kv(const float* __restrict__ kp, const float* __restrict__ v,
                     float* kvm) {
  int bh = blockIdx.y;
  int b = bh / HEADS, hh = bh % HEADS;
  int lane = threadIdx.x & 31, wave = threadIdx.x >> 5;
  int tile = blockIdx.x * 8 + wave;                    // 0..63
  int mt = tile >> 2, dt = tile & 3;
  int l16 = lane & 15, kh = (lane >> 4) << 1, hi8 = (lane >> 4) << 3;
  int m0 = mt * 16, d0 = dt * 16;
  v8f acc = {};
  for (int k0 = 0; k0 < TSEQ; k0 += 4) {
    float ax = kp[((size_t)bh * TSEQ + k0 + kh) * MFEAT + m0 + l16];
    float ay = kp[((size_t)bh * TSEQ + k0 + kh + 1) * MFEAT + m0 + l16];
    float bx = v[((size_t)(b * TSEQ + k0 + kh) * HEADS + hh) * DDIM + d0 + l16];
    float by = v[((size_t)(b * TSEQ + k0 + kh + 1) * HEADS + hh) * DDIM + d0 + l16];
    acc = wmma4(ax, ay, bx, by, acc);
  }
  for (int rr = 0; rr < 8; ++rr)
    kvm[((size_t)bh * MFEAT + m0 + rr + hi8) * DDIM + d0 + l16] = acc[rr];
}

// qkv[bh][t][d] = q'[t][:] @ kv[:,d]          (WMMA, K = 256)
__global__ void k_qkv(const float* __restrict__ qp, const float* __restrict__ kvm,
                      float* qkv) {
  int bh = blockIdx.y;
  int lane = threadIdx.x & 31, wave = threadIdx.x >> 5;
  int rt = wave >> 2, dt = wave & 3;
  int m0 = blockIdx.x * 32 + rt * 16, d0 = dt * 16;
  int l16 = lane & 15, kh = (lane >> 4) << 1, hi8 = (lane >> 4) << 3;
  v8f acc = {};
  for (int k0 = 0; k0 < MFEAT; k0 += 4) {
    float ax = qp[((size_t)bh * TSEQ + m0 + l16) * MFEAT + k0 + kh];
    float ay = qp[((size_t)bh * TSEQ + m0 + l16) * MFEAT + k0 + kh + 1];
    float bx = kvm[((size_t)bh * MFEAT + k0 + kh) * DDIM + d0 + l16];
    float by = kvm[((size_t)bh * MFEAT + k0 + kh + 1) * DDIM + d0 + l16];
    acc = wmma4(ax, ay, bx, by, acc);
  }
  for (int rr = 0; rr < 8; ++rr)
    qkv[((size_t)bh * TSEQ + m0 + rr + hi8) * DDIM + d0 + l16] = acc[rr];
}

__global__ void k_qk1(const float* __restrict__ qp, const float* __restrict__ ksum,
                      float* qk1) {
  int gid = blockIdx.x * blockDim.x + threadIdx.x;    // over BH*TSEQ
  int bh = gid / TSEQ;
  const float* row = qp + (size_t)gid * MFEAT;
  const float* ks = ksum + bh * MFEAT;
  float s = 0.f;
  for (int m = 0; m < MFEAT; ++m) s += row[m] * ks[m];
  qk1[gid] = s;
}

// pass 1: per bucket, inner = s_q s_k^T * temp - log(dup); lse = max(rowmax, pls)
__global__ void __launch_bounds__(256) k_pass1(
    const float* __restrict__ q, const float* __restrict__ k,
    const int* qpos, const int* kpos, const int* qrev, const int* krev,
    const float* pls, float* lse_g) {
  int n = blockIdx.x, bh = blockIdx.y, h = blockIdx.z;
  int b = bh / HEADS, hh = bh % HEADS;
  __shared__ int sqi[BKT], ski[BKT], sqb[NHASH][BKT], skb[NHASH][BKT];
  __shared__ float sspls[BKT];
  int tid = threadIdx.x;
  if (tid < BKT) {
    int qi = qpos[((size_t)h * BH + bh) * TSEQ + n * BKT + tid];
    int ki = kpos[((size_t)h * BH + bh) * TSEQ + n * BKT + tid];
    sqi[tid] = qi; ski[tid] = ki;
    sspls[tid] = pls[(size_t)bh * TSEQ + qi];
    for (int j = 0; j < NHASH; ++j) {
      sqb[j][tid] = qrev[((size_t)j * BH + bh) * TSEQ + qi] >> 7;
      skb[j][tid] = krev[((size_t)j * BH + bh) * TSEQ + ki] >> 7;
    }
  }
  __syncthreads();
  int lane = threadIdx.x & 31, wave = threadIdx.x >> 5;
  int l16 = lane & 15, kh = (lane >> 4) << 1, hi8 = (lane >> 4) << 3;
  int m0 = wave * 16;
  size_t abase = ((size_t)(b * TSEQ + sqi[m0 + l16]) * HEADS + hh) * EDIM;
  float rm[8];
  for (int rr = 0; rr < 8; ++rr) rm[rr] = -3.4e38f;
  for (int ct = 0; ct < 8; ++ct) {
    int n0 = ct * 16;
    size_t bbase = ((size_t)(b * TSEQ + ski[n0 + l16]) * HEADS + hh) * EDIM;
    v8f acc = {};
    for (int e0 = 0; e0 < EDIM; e0 += 4)
      acc = wmma4(q[abase + e0 + kh], q[abase + e0 + kh + 1],
                  k[bbase + e0 + kh], k[bbase + e0 + kh + 1], acc);
    for (int rr = 0; rr < 8; ++rr) {
      int row = m0 + rr + hi8, col = n0 + l16;
      int dup = 0;
      for (int j = 0; j < NHASH; ++j) dup += (sqb[j][row] == skb[j][col]);
      rm[rr] = fmaxf(rm[rr], acc[rr] * TEMP - logf((float)dup));
    }
  }
  for (int rr = 0; rr < 8; ++rr)
    for (int off = 1; off < 16; off <<= 1)
      rm[rr] = fmaxf(rm[rr], __shfl_xor(rm[rr], off, 32));
  if (l16 == 0) {
    for (int rr = 0; rr < 8; ++rr) {
      int row = m0 + rr + hi8;
      lse_g[((size_t)h * BH + bh) * TSEQ + sqi[row]] = fmaxf(rm[rr], sspls[row]);
    }
  }
}

// logsumexp over hashes -> probs; initialize accumulators with Performer term
__global__ void k_probs(const float* lse_g, const float* pls, const float* qk1,
                        const float* qkv, float* probs, float* outacc, float* normacc) {
  int gid = blockIdx.x * blockDim.x + threadIdx.x;    // over BH*TSEQ
  int bh = gid / TSEQ, t = gid % TSEQ;
  float l[NHASH], m = -3.4e38f;
  for (int j = 0; j < NHASH; ++j) {
    l[j] = lse_g[((size_t)j * BH + bh) * TSEQ + t];
    m = fmaxf(m, l[j]);
  }
  float s = 0.f;
  for (int j = 0; j < NHASH; ++j) s += expf(l[j] - m);
  float nls = m + logf(s);
  for (int j = 0; j < NHASH; ++j)
    probs[((size_t)j * BH + bh) * TSEQ + t] = expf(l[j] - nls);
  float ps = expf(pls[gid] - nls);
  normacc[gid] = qk1[gid] * ps;
  for (int d = 0; d < DDIM; ++d)
    outacc[(size_t)gid * DDIM + d] = qkv[(size_t)gid * DDIM + d] * ps;
}

// pass 2: full bucket attention, prob-scaled, atomically accumulated
__global__ void __launch_bounds__(256) k_pass2(
    const float* __restrict__ q, const float* __restrict__ k,
    const float* __restrict__ v,
    const float* __restrict__ qp, const float* __restrict__ kp,
    const int* qpos, const int* kpos, const int* qrev, const int* krev,
    const float* pls, const float* lse_g, const float* probs,
    float* outacc, float* normacc) {
  extern __shared__ char smemraw[];
  float* sdots = (float*)smemraw;                       // [BKT][BKT+1]
  int*   sqi   = (int*)(sdots + BKT * (BKT + 1));
  int*   ski   = sqi + BKT;
  int*   sqb   = ski + BKT;                             // [NHASH][BKT]
  int*   skb   = sqb + NHASH * BKT;                     // [NHASH][BKT]
  float* sspls = (float*)(skb + NHASH * BKT);
  float* slse  = sspls + BKT;
  float* sprob = slse + BKT;

  int n = blockIdx.x, bh = blockIdx.y, h = blockIdx.z;
  int b = bh / HEADS, hh = bh % HEADS;
  int tid = threadIdx.x;
  if (tid < BKT) {
    int qi = qpos[((size_t)h * BH + bh) * TSEQ + n * BKT + tid];
    int ki = kpos[((size_t)h * BH + bh) * TSEQ + n * BKT + tid];
    sqi[tid] = qi; ski[tid] = ki;
    sspls[tid] = pls[(size_t)bh * TSEQ + qi];
    slse[tid]  = lse_g[((size_t)h * BH + bh) * TSEQ + qi];
    sprob[tid] = probs[((size_t)h * BH + bh) * TSEQ + qi];
    for (int j = 0; j < NHASH; ++j) {
      sqb[j * BKT + tid] = qrev[((size_t)j * BH + bh) * TSEQ + qi] >> 7;
      skb[j * BKT + tid] = krev[((size_t)j * BH + bh) * TSEQ + ki] >> 7;
    }
  }
  __syncthreads();
  int lane = threadIdx.x & 31, wave = threadIdx.x >> 5;
  int l16 = lane & 15, kh = (lane >> 4) << 1, hi8 = (lane >> 4) << 3;
  int m0 = wave * 16;
  size_t abase  = ((size_t)(b * TSEQ + sqi[m0 + l16]) * HEADS + hh) * EDIM;
  size_t pabase = ((size_t)bh * TSEQ + sqi[m0 + l16]) * MFEAT;
  for (int ct = 0; ct < 8; ++ct) {
    int n0 = ct * 16;
    size_t bbase  = ((size_t)(b * TSEQ + ski[n0 + l16]) * HEADS + hh) * EDIM;
    size_t pbbase = ((size_t)bh * TSEQ + ski[n0 + l16]) * MFEAT;
    v8f ai = {};
    for (int e0 = 0; e0 < EDIM; e0 += 4)
      ai = wmma4(q[abase + e0 + kh], q[abase + e0 + kh + 1],
                 k[bbase + e0 + kh], k[bbase + e0 + kh + 1], ai);
    v8f ap = {};
    for (int e0 = 0; e0 < MFEAT; e0 += 4)
      ap = wmma4(qp[pabase + e0 + kh], qp[pabase + e0 + kh + 1],
                 kp[pbbase + e0 + kh], kp[pbbase + e0 + kh + 1], ap);
    for (int rr = 0; rr < 8; ++rr) {
      int row = m0 + rr + hi8, col = n0 + l16;
      int dup = 0;
      for (int j = 0; j < NHASH; ++j) dup += (sqb[j * BKT + row] == skb[j * BKT + col]);
      float fd = (float)dup;
      float innerv = ai[rr] * TEMP - logf(fd);
      float dpv = ap[rr] / fd;
      float lse = slse[row];
      float dv = (expf(innerv - lse) - dpv * expf(sspls[row] - lse)) * sprob[row];
      sdots[row * (BKT + 1) + col] = dv;
    }
  }
  __syncthreads();
  if (tid < BKT) {
    float s = 0.f;
    for (int j = 0; j < BKT; ++j) s += sdots[tid * (BKT + 1) + j];
    atomicAdd(&normacc[(size_t)bh * TSEQ + sqi[tid]], s);
  }
  // so = dots @ s_v  (A from LDS in A-fragment layout, B gathered from global)
  for (int dt = 0; dt < 4; ++dt) {
    int d0 = dt * 16;
    v8f acc = {};
    for (int j0 = 0; j0 < BKT; j0 += 4) {
      float ax = sdots[(m0 + l16) * (BKT + 1) + j0 + kh];
      float ay = sdots[(m0 + l16) * (BKT + 1) + j0 + kh + 1];
      float bx = v[((size_t)(b * TSEQ + ski[j0 + kh]) * HEADS + hh) * DDIM + d0 + l16];
      float by = v[((size_t)(b * TSEQ + ski[j0 + kh + 1]) * HEADS + hh) * DDIM + d0 + l16];
      acc = wmma4(ax, ay, bx, by, acc);
    }
    for (int rr = 0; rr < 8; ++rr) {
      int row = m0 + rr + hi8;
      atomicAdd(&outacc[((size_t)bh * TSEQ + sqi[row]) * DDIM + d0 + l16], acc[rr]);
    }
  }
}

__global__ void k_final(const float* outacc, const float* normacc, float* out) {
  size_t gid = (size_t)blockIdx.x * blockDim.x + threadIdx.x;  // BH*TSEQ*DDIM
  size_t bt = gid / DDIM;
  int d = (int)(gid % DDIM);
  int bh = (int)(bt / TSEQ), t = (int)(bt % TSEQ);
  int b = bh / HEADS, hh = bh % HEADS;
  float nn = fmaxf(normacc[bt], 1e-6f);
  out[((size_t)(b * TSEQ + t) * HEADS + hh) * DDIM + d] = outacc[gid] / nn;
}

// ---------------------------------------------------------------------------
extern "C" void kernel_launch(void* const* d_in, const int* in_sizes, int n_in,
                              void* d_out, int out_size, void* d_ws, size_t ws_size,
                              hipStream_t stream) {
  (void)in_sizes; (void)n_in; (void)out_size; (void)ws_size;
  const float* q     = (const float*)d_in[0];
  const float* k     = (const float*)d_in[1];
  const float* v     = (const float*)d_in[2];
  const float* proj  = (const float*)d_in[3];
  const float* alpha = (const float*)d_in[4];
  const float* beta  = (const float*)d_in[5];
  float* out = (float*)d_out;

  char* w = (char*)d_ws;
  auto bump = [&](size_t elems) -> void* {
    void* p = (void*)w;
    w += ((elems * 4 + 255) & ~(size_t)255);
    return p;
  };
  float*    qn2     = (float*)bump((size_t)BH * TSEQ);
  float*    kn2     = (float*)bump((size_t)BH * TSEQ);
  unsigned* mq2     = (unsigned*)bump(BH);
  unsigned* mk2     = (unsigned*)bump(BH);
  unsigned* kmaxb   = (unsigned*)bump(BH);
  float*    qhash   = (float*)bump((size_t)NHASH * BH * TSEQ);
  float*    khash   = (float*)bump((size_t)NHASH * BH * TSEQ);
  int*      qpos    = (int*)bump((size_t)NHASH * BH * TSEQ);
  int*      kpos    = (int*)bump((size_t)NHASH * BH * TSEQ);
  int*      qrev    = (int*)bump((size_t)NHASH * BH * TSEQ);
  int*      krev    = (int*)bump((size_t)NHASH * BH * TSEQ);
  float*    qprime  = (float*)bump((size_t)BH * TSEQ * MFEAT);  // logv_q -> q'
  float*    kprime  = (float*)bump((size_t)BH * TSEQ * MFEAT);  // logv_k -> k'
  float*    qls     = (float*)bump((size_t)BH * TSEQ);
  float*    pls     = (float*)bump((size_t)BH * TSEQ);
  float*    kvm     = (float*)bump((size_t)BH * MFEAT * DDIM);
  float*    ksum    = (float*)bump((size_t)BH * MFEAT);
  float*    qkv     = (float*)bump((size_t)BH * TSEQ * DDIM);
  float*    qk1     = (float*)bump((size_t)BH * TSEQ);
  float*    lse_g   = (float*)bump((size_t)NHASH * BH * TSEQ);
  float*    probs   = (float*)bump((size_t)NHASH * BH * TSEQ);
  float*    outacc  = (float*)bump((size_t)BH * TSEQ * DDIM);
  float*    normacc = (float*)bump((size_t)BH * TSEQ);

  hipLaunchKernelGGL(k_init, dim3(1), dim3(64), 0, stream, mq2, mk2, kmaxb);
  hipLaunchKernelGGL(k_norms, dim3(BH * TSEQ / 256), dim3(256), 0, stream,
                     q, k, qn2, kn2, mq2, mk2);
  hipLaunchKernelGGL(k_hash, dim3(BH * TSEQ / 256), dim3(256), 0, stream,
                     q, k, alpha, beta, qn2, kn2, mq2, mk2, qhash, khash);
  hipLaunchKernelGGL(k_sort, dim3(NHASH * BH), dim3(512), 0, stream, qhash, qpos, qrev);
  hipLaunchKernelGGL(k_sort, dim3(NHASH * BH), dim3(512), 0, stream, khash, kpos, krev);
  hipLaunchKernelGGL(k_logv, dim3(TSEQ / 16, BH, 2), dim3(512), 0, stream,
                     q, k, proj, qn2, kn2, qprime, kprime, kmaxb);
  hipLaunchKernelGGL(k_qls, dim3(BH * TSEQ / 256), dim3(256), 0, stream,
                     qprime, kmaxb, qls, pls);
  hipLaunchKernelGGL(k_prime, dim3((unsigned)((size_t)BH * TSEQ * MFEAT / 256)),
                     dim3(256), 0, stream, qprime, kprime, qls, kmaxb);
  hipLaunchKernelGGL(k_ksum, dim3(BH), dim3(MFEAT), 0, stream, kprime, ksum);
  hipLaunchKernelGGL(k_kv, dim3(8, BH), dim3(256), 0, stream, kprime, v, kvm);
  hipLaunchKernelGGL(k_qkv, dim3(TSEQ / 32, BH), dim3(256), 0, stream, qprime, kvm, qkv);
  hipLaunchKernelGGL(k_qk1, dim3(BH * TSEQ / 256), dim3(256), 0, stream,
                     qprime, ksum, qk1);
  hipLaunchKernelGGL(k_pass1, dim3(NB, BH, NHASH), dim3(256), 0, stream,
                     q, k, qpos, kpos, qrev, krev, pls, lse_g);
  hipLaunchKernelGGL(k_probs, dim3(BH * TSEQ / 256), dim3(256), 0, stream,
                     lse_g, pls, qk1, qkv, probs, outacc, normacc);
  size_t p2_smem = (size_t)BKT * (BKT + 1) * 4   // sdots
                 + (size_t)(2 + 2 * NHASH) * BKT * 4  // sqi/ski/sqb/skb
                 + (size_t)3 * BKT * 4;          // sspls/slse/sprob
  hipLaunchKernelGGL(k_pass2, dim3(NB, BH, NHASH), dim3(256), p2_smem, stream,
                     q, k, v, qprime, kprime, qpos, kpos, qrev, krev,
                     pls, lse_g, probs, outacc, normacc);
  hipLaunchKernelGGL(k_final, dim3((unsigned)((size_t)BH * TSEQ * DDIM / 256)),
                     dim3(256), 0, stream, outacc, normacc, out);
}